// MMGATLayer_9801115369730
// MI455X (gfx1250) — compile-verified
//
#include <hip/hip_runtime.h>
#include <cstdint>
#include <cstddef>

typedef __attribute__((ext_vector_type(2))) float v2f;
typedef __attribute__((ext_vector_type(8))) float v8f;

#define D128 128
#define NEG_SLOPE 0.2f
#define LN_EPS 1e-5f

// ---- monotone float <-> uint32 key (for atomicMax on floats of any sign) ----
__device__ __forceinline__ unsigned fkey_encode(float f) {
  unsigned u = __float_as_uint(f);
  return u ^ (unsigned)(((int)u >> 31) | 0x80000000);
}
__device__ __forceinline__ float fkey_decode(unsigned u) {
  return (u & 0x80000000u) ? __uint_as_float(u ^ 0x80000000u)
                           : __uint_as_float(~u);
}

// ------------------------------------------------------------------
// Kernel 0: init accumulators (out, emax, denom) every call
// ------------------------------------------------------------------
__global__ void gat_init_kernel(float* __restrict__ out,
                                unsigned* __restrict__ emax,
                                float* __restrict__ denom, int N) {
  int i = blockIdx.x * blockDim.x + threadIdx.x;
  if (i < N * D128) out[i] = 0.0f;
  if (i < N) {
    emax[i]  = fkey_encode(-__builtin_inff());
    denom[i] = 0.0f;
  }
}

// ------------------------------------------------------------------
// Kernel 1: x_l = x@W_l, x_r = x@W_r via V_WMMA_F32_16X16X4_F32.
// Block = 256 threads = 8 waves; each wave owns one 16x16 output tile
// (wave -> column tile 0..7), block -> 16-row slab. W staged in LDS.
// blockIdx.y selects (W_l -> x_l) or (W_r -> x_r).
// ------------------------------------------------------------------
__global__ __launch_bounds__(256)
void gat_gemm_wmma_kernel(const float* __restrict__ x,
                          const float* __restrict__ Wl,
                          const float* __restrict__ Wr,
                          float* __restrict__ xl,
                          float* __restrict__ xr, int nrows) {
  __shared__ float sW[D128 * D128];  // 64 KB (WGP has 320 KB)
  const float* W  = (blockIdx.y == 0) ? Wl : Wr;
  float*      out = (blockIdx.y == 0) ? xl : xr;

  // cooperative coalesced copy of W into LDS
  {
    const float4* s4 = (const float4*)W;
    float4*       d4 = (float4*)sW;
    for (int i = threadIdx.x; i < (D128 * D128) / 4; i += 256) d4[i] = s4[i];
  }
  __syncthreads();

  const int wave  = threadIdx.x >> 5;
  const int lane  = threadIdx.x & 31;
  const int l16   = lane & 15;
  const int khalf = lane >> 4;             // 0: K={0,1}, 1: K={2,3}
  const int rowt  = blockIdx.x * 16;
  if (rowt >= nrows) return;
  const int colt  = wave * 16;
  const int row   = rowt + l16;            // A: M index
  const int col   = colt + l16;            // B/C/D: N index

  v8f acc = {};
  for (int k = 0; k < D128; k += 4) {
    const int ka = k + khalf * 2;
    // A fragment (16x4 f32): 2 consecutive K per lane -> one b64 load
    float2 av = *(const float2*)(x + (size_t)row * D128 + ka);
    v2f a; a.x = av.x; a.y = av.y;
    // B fragment (4x16 f32) from LDS, row-major W
    v2f b; b.x = sW[ka * D128 + col];
           b.y = sW[(ka + 1) * D128 + col];
    acc = __builtin_amdgcn_wmma_f32_16x16x4_f32(false, a, false, b,
                                                (short)0, acc, false, false);
  }
  // C/D layout: VGPR v -> M = v + khalf*8, N = col
#pragma unroll
  for (int v = 0; v < 8; ++v) {
    out[(size_t)(rowt + v + khalf * 8) * D128 + col] = acc[v];
  }
}

// ------------------------------------------------------------------
// Kernel 2: per-edge GATv2 score e = att . leakyrelu(x_l[src]+x_r[dst])
// One wave32 per edge, 4 dims/lane (b128 loads), shuffle reduction,
// atomic segment-max via monotone-uint key.
// ------------------------------------------------------------------
__global__ __launch_bounds__(256)
void gat_edge_score_kernel(const float* __restrict__ xl,
                           const float* __restrict__ xr,
                           const float* __restrict__ att,
                           const int64_t* __restrict__ ei, int E, int Etot,
                           float* __restrict__ ebuf,
                           unsigned* __restrict__ emax) {
  int edge = blockIdx.x * 8 + (threadIdx.x >> 5);
  if (edge >= Etot) return;
  int lane = threadIdx.x & 31;
  int64_t src, dst;
  if (edge < E) { src = ei[edge]; dst = ei[(size_t)E + edge]; }
  else          { src = dst = edge - E; }

  const float4 a = *(const float4*)(att + lane * 4);
  const float4 l = *(const float4*)(xl + (size_t)src * D128 + lane * 4);
  const float4 r = *(const float4*)(xr + (size_t)dst * D128 + lane * 4);

  float s = 0.0f, h;
  h = l.x + r.x; h = h > 0.0f ? h : NEG_SLOPE * h; s += a.x * h;
  h = l.y + r.y; h = h > 0.0f ? h : NEG_SLOPE * h; s += a.y * h;
  h = l.z + r.z; h = h > 0.0f ? h : NEG_SLOPE * h; s += a.z * h;
  h = l.w + r.w; h = h > 0.0f ? h : NEG_SLOPE * h; s += a.w * h;
#pragma unroll
  for (int off = 16; off > 0; off >>= 1) s += __shfl_xor(s, off, 32);

  if (lane == 0) {
    ebuf[edge] = s;
    atomicMax(&emax[dst], fkey_encode(s));
  }
}

// ------------------------------------------------------------------
// Kernel 3: ex = exp(e - max[dst]); denom[dst] += ex   (one thread/edge)
// ------------------------------------------------------------------
__global__ void gat_edge_exp_kernel(const int64_t* __restrict__ ei, int E,
                                    int Etot, float* __restrict__ ebuf,
                                    const unsigned* __restrict__ emax,
                                    float* __restrict__ denom) {
  int edge = blockIdx.x * blockDim.x + threadIdx.x;
  if (edge >= Etot) return;
  int64_t dst = (edge < E) ? ei[(size_t)E + edge] : (int64_t)(edge - E);
  float ex = expf(ebuf[edge] - fkey_decode(emax[dst]));
  ebuf[edge] = ex;
  atomicAdd(&denom[dst], ex);
}

// ------------------------------------------------------------------
// Kernel 4: out[dst] += (ex/denom[dst]) * x_l[src]  (wave per edge,
// 4 global_atomic_add_f32 per lane; rows are L2-resident)
// ------------------------------------------------------------------
__global__ __launch_bounds__(256)
void gat_aggregate_kernel(const float* __restrict__ xl,
                          const int64_t* __restrict__ ei, int E, int Etot,
                          const float* __restrict__ ebuf,
                          const float* __restrict__ denom,
                          float* __restrict__ out) {
  int edge = blockIdx.x * 8 + (threadIdx.x >> 5);
  if (edge >= Etot) return;
  int lane = threadIdx.x & 31;
  int64_t src, dst;
  if (edge < E) { src = ei[edge]; dst = ei[(size_t)E + edge]; }
  else          { src = dst = edge - E; }

  float alpha    = ebuf[edge] / denom[dst];
  const float4 l = *(const float4*)(xl + (size_t)src * D128 + lane * 4);
  float* o       = out + (size_t)dst * D128 + lane * 4;
  atomicAdd(o + 0, alpha * l.x);
  atomicAdd(o + 1, alpha * l.y);
  atomicAdd(o + 2, alpha * l.z);
  atomicAdd(o + 3, alpha * l.w);
}

// ------------------------------------------------------------------
// Kernel 5: out = LayerNorm(out + bias) * gamma + beta  (wave per row)
// ------------------------------------------------------------------
__global__ __launch_bounds__(256)
void gat_layernorm_kernel(float* __restrict__ out,
                          const float* __restrict__ bias,
                          const float* __restrict__ gamma,
                          const float* __restrict__ beta, int N) {
  int node = blockIdx.x * 8 + (threadIdx.x >> 5);
  if (node >= N) return;
  int lane   = threadIdx.x & 31;
  float* row = out + (size_t)node * D128 + lane * 4;

  float4 v       = *(const float4*)row;
  const float4 b = *(const float4*)(bias + lane * 4);
  v.x += b.x; v.y += b.y; v.z += b.z; v.w += b.w;

  float s = v.x + v.y + v.z + v.w;
#pragma unroll
  for (int off = 16; off > 0; off >>= 1) s += __shfl_xor(s, off, 32);
  float mu = s * (1.0f / 128.0f);

  float dx = v.x - mu, dy = v.y - mu, dz = v.z - mu, dw = v.w - mu;
  float q = dx * dx + dy * dy + dz * dz + dw * dw;
#pragma unroll
  for (int off = 16; off > 0; off >>= 1) q += __shfl_xor(q, off, 32);
  float inv = rsqrtf(q * (1.0f / 128.0f) + LN_EPS);

  const float4 g  = *(const float4*)(gamma + lane * 4);
  const float4 bt = *(const float4*)(beta + lane * 4);
  float4 o;
  o.x = dx * inv * g.x + bt.x;
  o.y = dy * inv * g.y + bt.y;
  o.z = dz * inv * g.z + bt.z;
  o.w = dw * inv * g.w + bt.w;
  *(float4*)row = o;
}

// ------------------------------------------------------------------
extern "C" void kernel_launch(void* const* d_in, const int* in_sizes, int n_in,
                              void* d_out, int out_size, void* d_ws,
                              size_t ws_size, hipStream_t stream) {
  const float*   x     = (const float*)d_in[0];
  const int64_t* ei    = (const int64_t*)d_in[1];
  const float*   Wl    = (const float*)d_in[2];
  const float*   Wr    = (const float*)d_in[3];
  const float*   att   = (const float*)d_in[4];
  const float*   bias  = (const float*)d_in[5];
  const float*   gamma = (const float*)d_in[6];
  const float*   beta  = (const float*)d_in[7];
  float*         out   = (float*)d_out;

  const int N    = in_sizes[0] / D128;   // 50000
  const int E    = in_sizes[1] / 2;      // 600000
  const int Etot = E + N;                // edges + self loops

  // carve workspace (all 16B aligned)
  char* ws = (char*)d_ws;
  float* xl = (float*)ws;     ws += (size_t)N * D128 * sizeof(float);
  float* xr = (float*)ws;     ws += (size_t)N * D128 * sizeof(float);
  float* ebuf = (float*)ws;   ws += ((size_t)Etot + 3ull) / 4ull * 4ull * sizeof(float);
  unsigned* emax = (unsigned*)ws; ws += (size_t)N * sizeof(unsigned);
  float* denom = (float*)ws;  ws += (size_t)N * sizeof(float);

  // 0) init accumulators
  gat_init_kernel<<<(N * D128 + 255) / 256, 256, 0, stream>>>(out, emax, denom, N);

  // 1) WMMA GEMMs: x_l = x@W_l, x_r = x@W_r
  dim3 ggrid((N + 15) / 16, 2);
  gat_gemm_wmma_kernel<<<ggrid, 256, 0, stream>>>(x, Wl, Wr, xl, xr, N);

  // 2) edge scores + segment max
  gat_edge_score_kernel<<<(Etot + 7) / 8, 256, 0, stream>>>(xl, xr, att, ei, E,
                                                            Etot, ebuf, emax);
  // 3) exp + segment sum
  gat_edge_exp_kernel<<<(Etot + 255) / 256, 256, 0, stream>>>(ei, E, Etot, ebuf,
                                                              emax, denom);
  // 4) weighted scatter-add aggregation
  gat_aggregate_kernel<<<(Etot + 7) / 8, 256, 0, stream>>>(xl, ei, E, Etot,
                                                           ebuf, denom, out);
  // 5) bias + LayerNorm (in place on d_out)
  gat_layernorm_kernel<<<(N + 7) / 8, 256, 0, stream>>>(out, bias, gamma, beta, N);
}